// ATTNLTE_7645041787122
// MI455X (gfx1250) — compile-verified
//
#include <hip/hip_runtime.h>
#include <math.h>

typedef __attribute__((ext_vector_type(2))) float v2f;
typedef __attribute__((ext_vector_type(8))) float v8f;
typedef __attribute__((ext_vector_type(4))) unsigned int u32x4;
typedef __attribute__((ext_vector_type(8))) int i32x8;
typedef __attribute__((ext_vector_type(4))) int i32x4;

#define PI_F 3.14159265358979323846f

#define B_    2
#define H_    64
#define W_    64
#define Q_    16384
#define CENC  64
#define HID   256
#define BQ    (B_*Q_)        // 32768
#define M4    (4*BQ)         // 131072
#define NPIX  (B_*H_*W_)     // 8192
#define KCONV (CENC*9)       // 576
#define LDX   264            // padded 260

// LDS layout for GEMM (single shared block so TDM byte offsets are fixed):
//   As: 64 rows x 17 dwords  (A tile [m][k], stride 17, conflict-free)
//   Ws: 16 rows x 65 dwords  (W tile [k][n], stride 65)
#define AS_DW   (64*17)          // 1088 dwords
#define WS_BASE AS_DW            // dword offset of Ws
#define WS_BYTE (AS_DW*4)        // 4352 bytes
#define LDS_DW  (AS_DW + 16*65)  // 2128 dwords

// ------------------------------------------------ TDM 2D tile load -> LDS
// tile_w dwords per row, tile_h rows, row stride stride_dw (dwords),
// pad_interval: 3 -> pad 1 dword every 16 dwords; 5 -> every 64 dwords.
__device__ __forceinline__ void tdm_load_2d(unsigned lds_byte, const float* gsrc,
                                            unsigned tile_w, unsigned tile_h,
                                            unsigned stride_dw, unsigned pad_interval) {
    unsigned long long ga = (unsigned long long)(uintptr_t)gsrc;
    u32x4 g0;
    g0.x = 1u;                                        // count=1 (valid descriptor)
    g0.y = lds_byte;                                  // LDS byte address
    g0.z = (unsigned)(ga & 0xFFFFFFFFu);              // global addr [95:64]
    g0.w = (unsigned)(ga >> 32) | (2u << 30);         // addr hi | type=2 ("image")
    i32x8 g1;
    g1[0] = (int)((2u << 16) | (1u << 20) | (pad_interval << 22)); // 4B elems, pad_en, pad_amount=0(=1dw)
    g1[1] = (int)((tile_w & 0xFFFFu) << 16);          // tensor_dim0 = tile_w (lo16)
    g1[2] = (int)((tile_w >> 16) | ((tile_h & 0xFFFFu) << 16));    // dim0 hi | tensor_dim1 lo
    g1[3] = (int)((tile_h >> 16) | (tile_w << 16));   // dim1 hi | tile_dim0
    g1[4] = (int)tile_h;                              // tile_dim1 | tile_dim2=0
    g1[5] = (int)stride_dw;                           // tensor_dim0_stride lo32
    g1[6] = 0;                                        // stride hi | dim1_stride lo
    g1[7] = 0;
    i32x4 z = {0, 0, 0, 0};
#if defined(__clang_major__) && (__clang_major__ >= 23)
    i32x8 z8 = {0, 0, 0, 0, 0, 0, 0, 0};
    __builtin_amdgcn_tensor_load_to_lds(g0, g1, z, z, z8, 0);
#else
    __builtin_amdgcn_tensor_load_to_lds(g0, g1, z, z, 0);
#endif
}

// ---------------------------------------------------------------- conv 3->64
__global__ void k_conv_enc(const float* __restrict__ inp, const float* __restrict__ w,
                           const float* __restrict__ bias, float* __restrict__ feat) {
    int idx = blockIdx.x * blockDim.x + threadIdx.x;      // B*CENC*H*W = 2^19
    if (idx >= B_ * CENC * H_ * W_) return;
    int x = idx & 63, y = (idx >> 6) & 63, o = (idx >> 12) & 63, bb = idx >> 18;
    float acc = bias[o];
    for (int c = 0; c < 3; ++c) {
        const float* ip = inp + ((bb * 3 + c) << 12);
        const float* wp = w + (o * 3 + c) * 9;
#pragma unroll
        for (int dy = 0; dy < 3; ++dy) {
            int yy = y + dy - 1;
            if (yy < 0 || yy > 63) continue;
#pragma unroll
            for (int dx = 0; dx < 3; ++dx) {
                int xx = x + dx - 1;
                if (xx < 0 || xx > 63) continue;
                acc += ip[(yy << 6) + xx] * wp[dy * 3 + dx];
            }
        }
    }
    feat[idx] = acc;
}

// ------------------------------------------- conv weights (O,C,3,3) -> (K,O)
__global__ void k_wt(const float* __restrict__ cw, const float* __restrict__ fw,
                     float* __restrict__ ct, float* __restrict__ ft) {
    int idx = blockIdx.x * blockDim.x + threadIdx.x;      // KCONV*HID
    if (idx >= KCONV * HID) return;
    int o = idx % HID, k = idx / HID;
    ct[idx] = cw[o * KCONV + k];
    ft[idx] = fw[o * KCONV + k];
}

// -------------------------------------------------------------------- im2col
__global__ void k_im2col(const float* __restrict__ feat, float* __restrict__ col) {
    int idx = blockIdx.x * blockDim.x + threadIdx.x;      // NPIX*KCONV
    if (idx >= NPIX * KCONV) return;
    int k = idx % KCONV, pix = idx / KCONV;
    int x = pix & 63, y = (pix >> 6) & 63, bb = pix >> 12;
    int dx = k % 3, dy = (k / 3) % 3, c = k / 9;
    int yy = y + dy - 1, xx = x + dx - 1;
    float v = 0.f;
    if (yy >= 0 && yy < 64 && xx >= 0 && xx < 64)
        v = feat[((bb * CENC + c) << 12) + (yy << 6) + xx];
    col[idx] = v;
}

// ------------------------------------------------- generic fp32 WMMA GEMM
// C[M,N] = act(A[M,K] @ W[K,N] + bias) (* mul)
// block: 128 thr (4 waves), block tile 64x64, wave tile 32x32 (2x2 frags)
__device__ __forceinline__ void gemm_chunk(const float* lds, v8f acc[2][2],
                                           int wm, int wn, int half, int l) {
#pragma unroll
    for (int kk = 0; kk < 16; kk += 4) {
        v2f a[2], b[2];
#pragma unroll
        for (int t = 0; t < 2; ++t) {
            int arow = wm + t * 16 + l;
            a[t].x = lds[arow * 17 + kk + 2 * half];
            a[t].y = lds[arow * 17 + kk + 2 * half + 1];
            int bcol = wn + t * 16 + l;
            b[t].x = lds[WS_BASE + (kk + 2 * half) * 65 + bcol];
            b[t].y = lds[WS_BASE + (kk + 2 * half + 1) * 65 + bcol];
        }
#pragma unroll
        for (int ti = 0; ti < 2; ++ti)
#pragma unroll
            for (int tj = 0; tj < 2; ++tj)
                acc[ti][tj] = __builtin_amdgcn_wmma_f32_16x16x4_f32(
                    false, a[ti], false, b[tj], (short)0, acc[ti][tj], false, false);
    }
}

__global__ __launch_bounds__(128)
void k_gemm(const float* __restrict__ A, int lda,
            const float* __restrict__ Wm, int ldw,
            const float* __restrict__ bias,
            float* __restrict__ C, int ldc,
            const float* __restrict__ mul, int ldm,
            int K, int relu) {
    __shared__ float lds[LDS_DW];
    int tid  = threadIdx.x;
    int lane = tid & 31, wid = tid >> 5;
    int m0 = blockIdx.y << 6, n0 = blockIdx.x << 6;
    int wm = (wid >> 1) << 5, wn = (wid & 1) << 5;
    int half = lane >> 4, l = lane & 15;
    v8f acc[2][2] = {};

    int kmain = K & ~15;
    for (int k0 = 0; k0 < kmain; k0 += 16) {
        if (wid == 0) {                                // TDM DMA: global -> LDS
            tdm_load_2d(0u, A + (long)m0 * lda + k0, 16u, 64u, (unsigned)lda, 3u);
            tdm_load_2d(WS_BYTE, Wm + (long)k0 * ldw + n0, 64u, 16u, (unsigned)ldw, 5u);
            __builtin_amdgcn_s_wait_tensorcnt(0);
        }
        __syncthreads();
        gemm_chunk(lds, acc, wm, wn, half, l);
        __syncthreads();
    }
    if (kmain < K) {                                   // single guarded tail chunk
        int kr = K - kmain;
#pragma unroll
        for (int i = 0; i < 8; ++i) {
            int e = tid + (i << 7);
            int r = e >> 4, c = e & 15;
            lds[r * 17 + c] = (c < kr) ? A[(long)(m0 + r) * lda + kmain + c] : 0.f;
        }
#pragma unroll
        for (int i = 0; i < 8; ++i) {
            int e = tid + (i << 7);
            int r = e >> 6, c = e & 63;
            lds[WS_BASE + r * 65 + c] = (r < kr) ? Wm[(long)(kmain + r) * ldw + n0 + c] : 0.f;
        }
        __syncthreads();
        gemm_chunk(lds, acc, wm, wn, half, l);
    }

#pragma unroll
    for (int ti = 0; ti < 2; ++ti) {
#pragma unroll
        for (int tj = 0; tj < 2; ++tj) {
            int cg = n0 + wn + tj * 16 + l;
            float bv = bias ? bias[cg] : 0.f;
#pragma unroll
            for (int i = 0; i < 8; ++i) {
                int rg = m0 + wm + ti * 16 + i + half * 8;
                float v = acc[ti][tj][i] + bv;
                if (relu) v = fmaxf(v, 0.f);
                if (mul)  v *= mul[(long)rg * ldm + cg];
                C[(long)rg * ldc + cg] = v;
            }
        }
    }
}

// ----------------------------------------------- feature map: coef * enc(freq)
__global__ void k_fmap(const float* __restrict__ coefN, const float* __restrict__ freqN,
                       float* __restrict__ fmN) {
    int pix = blockIdx.x, j = threadIdx.x;            // 8192 x 128
    const float* fr = freqN + pix * HID;
    float f = fr[2 * j] + fr[2 * j + 1];
    float sn, cs;
    sincosf(PI_F * f, &sn, &cs);
    fmN[pix * HID + j]       = coefN[pix * HID + j] * cs;
    fmN[pix * HID + 128 + j] = coefN[pix * HID + 128 + j] * sn;
}

// ------------------------------- per-(shift,query) kv + MLP-input row (260)
__global__ void k_prepare(const float* __restrict__ coord, const float* __restrict__ cell,
                          const float* __restrict__ coefN, const float* __restrict__ freqN,
                          const float* __restrict__ phase_w, float* __restrict__ X) {
    int r = blockIdx.x, j = threadIdx.x;              // M4 blocks x 128
    int s  = r >> 15;                                 // r / BQ
    int bq = r & (BQ - 1);
    int b  = bq >> 14;                                // bq / Q_
    float cy = coord[bq * 2], cx = coord[bq * 2 + 1];
    float c00 = cell[(b * Q_) * 2], c01 = cell[(b * Q_) * 2 + 1];
    float rxs = (1.f - c00) / (H_ - 1.f);
    float rys = (1.f - c01) / (W_ - 1.f);
    float vx = (s & 2) ? 1.f : -1.f;
    float vy = (s & 1) ? 1.f : -1.f;
    float cy_ = fminf(fmaxf(cy + vx * rxs + 1e-6f, -1.f + 1e-6f), 1.f - 1e-6f);
    float cx_ = fminf(fmaxf(cx + vy * rys + 1e-6f, -1.f + 1e-6f), 1.f - 1e-6f);
    int iy = (int)fminf(fmaxf(rintf((cy_ + 1.f) * 32.f - 0.5f), 0.f), 63.f);
    int ix = (int)fminf(fmaxf(rintf((cx_ + 1.f) * 32.f - 0.5f), 0.f), 63.f);
    int pix = (b << 12) + (iy << 6) + ix;
    float qcy = -1.f + (2.f * iy + 1.f) / 64.f;
    float qcx = -1.f + (2.f * ix + 1.f) / 64.f;
    float rel0 = (cy - qcy) * 64.f;
    float rel1 = (cx - qcx) * 64.f;
    float rc0 = cell[bq * 2] * 64.f, rc1 = cell[bq * 2 + 1] * 64.f;
    float fq0 = freqN[pix * HID + 2 * j], fq1 = freqN[pix * HID + 2 * j + 1];
    float ph  = rc0 * phase_w[2 * j] + rc1 * phase_w[2 * j + 1];
    float f = rel0 * fq0 + rel1 * fq1 + ph;
    float sn, cs;
    sincosf(PI_F * f, &sn, &cs);
    long xo = (long)r * LDX;
    X[xo + j]       = coefN[pix * HID + j] * cs;
    X[xo + 128 + j] = coefN[pix * HID + 128 + j] * sn;
    if (j == 0) {
        X[xo + 256] = rel0; X[xo + 257] = rel1;
        X[xo + 258] = rc0;  X[xo + 259] = rc1;
    }
}

// --------------------------------------- attention over 4 shifts, wave/query
__global__ void k_attn(const float* __restrict__ coord, const float* __restrict__ fmN,
                       const float* __restrict__ Pk, const float* __restrict__ Pv,
                       float* __restrict__ Xattn) {
    int wid = threadIdx.x >> 5, lane = threadIdx.x & 31;
    int bq = blockIdx.x * 8 + wid;
    int b = bq >> 14;
    float cy = coord[bq * 2], cx = coord[bq * 2 + 1];
    int iy = (int)fminf(fmaxf(rintf((cy + 1.f) * 32.f - 0.5f), 0.f), 63.f);
    int ix = (int)fminf(fmaxf(rintf((cx + 1.f) * 32.f - 0.5f), 0.f), 63.f);
    int pix = (b << 12) + (iy << 6) + ix;
    float q[8], lg[4];
#pragma unroll
    for (int i = 0; i < 8; ++i) q[i] = fmN[pix * HID + lane + i * 32];
#pragma unroll
    for (int s = 0; s < 4; ++s) {
        const float* p = Pk + (long)(s * BQ + bq) * HID;
        float d = 0.f;
#pragma unroll
        for (int i = 0; i < 8; ++i) d += q[i] * p[lane + i * 32];
        lg[s] = d;
    }
#pragma unroll
    for (int m = 16; m; m >>= 1) {
#pragma unroll
        for (int s = 0; s < 4; ++s) lg[s] += __shfl_xor(lg[s], m, 32);
    }
    float mx = fmaxf(fmaxf(lg[0], lg[1]), fmaxf(lg[2], lg[3]));
    float e[4], sum = 0.f;
#pragma unroll
    for (int s = 0; s < 4; ++s) { e[s] = expf(lg[s] - mx); sum += e[s]; }
    float inv = 1.f / sum;
#pragma unroll
    for (int i = 0; i < 8; ++i) {
        float x = 0.f;
#pragma unroll
        for (int s = 0; s < 4; ++s)
            x += e[s] * inv * Pv[(long)(s * BQ + bq) * HID + lane + i * 32];
        Xattn[(long)bq * HID + lane + i * 32] = x;
    }
}

// ------------------------------ final 256->3 projection + bilinear residual
__global__ void k_final(const float* __restrict__ Hq, const float* __restrict__ qw2,
                        const float* __restrict__ qb2, const float* __restrict__ inp,
                        const float* __restrict__ coord, float* __restrict__ out) {
    int bq = blockIdx.x * blockDim.x + threadIdx.x;
    if (bq >= BQ) return;
    int b = bq >> 14;
    const float* h = Hq + (long)bq * HID;
    float a0 = 0.f, a1 = 0.f, a2 = 0.f;
    for (int c = 0; c < HID; ++c) {
        float hv = h[c];
        a0 += hv * qw2[c * 3];
        a1 += hv * qw2[c * 3 + 1];
        a2 += hv * qw2[c * 3 + 2];
    }
    float cy = coord[bq * 2], cx = coord[bq * 2 + 1];
    float y = fminf(fmaxf((cy + 1.f) * 32.f - 0.5f, 0.f), 63.f);
    float x = fminf(fmaxf((cx + 1.f) * 32.f - 0.5f, 0.f), 63.f);
    float y0f = floorf(y), x0f = floorf(x);
    float wy = y - y0f, wx = x - x0f;
    int y0 = (int)y0f, x0 = (int)x0f;
    int y1 = min(y0 + 1, 63), x1 = min(x0 + 1, 63);
    const float* ip = inp + b * 3 * 4096;
    float acc[3] = {a0, a1, a2};
#pragma unroll
    for (int ch = 0; ch < 3; ++ch) {
        const float* c0 = ip + ch * 4096;
        float v00 = c0[(y0 << 6) + x0], v01 = c0[(y0 << 6) + x1];
        float v10 = c0[(y1 << 6) + x0], v11 = c0[(y1 << 6) + x1];
        float bi = v00 * (1.f - wy) * (1.f - wx) + v01 * (1.f - wy) * wx
                 + v10 * wy * (1.f - wx)         + v11 * wy * wx;
        out[bq * 3 + ch] = acc[ch] + qb2[ch] + bi;
    }
}

extern "C" void kernel_launch(void* const* d_in, const int* in_sizes, int n_in,
                              void* d_out, int out_size, void* d_ws, size_t ws_size,
                              hipStream_t stream) {
    const float* inp     = (const float*)d_in[0];
    const float* coord   = (const float*)d_in[1];
    const float* cell    = (const float*)d_in[2];
    const float* enc_w   = (const float*)d_in[3];
    const float* enc_b   = (const float*)d_in[4];
    const float* coef_w  = (const float*)d_in[5];
    const float* coef_b  = (const float*)d_in[6];
    const float* freq_w  = (const float*)d_in[7];
    const float* freq_b  = (const float*)d_in[8];
    const float* phase_w = (const float*)d_in[9];
    const float* kw1 = (const float*)d_in[10]; const float* kb1 = (const float*)d_in[11];
    const float* kw2 = (const float*)d_in[12]; const float* kb2 = (const float*)d_in[13];
    const float* vw1 = (const float*)d_in[14]; const float* vb1 = (const float*)d_in[15];
    const float* vw2 = (const float*)d_in[16]; const float* vb2 = (const float*)d_in[17];
    const float* qw1 = (const float*)d_in[18]; const float* qb1 = (const float*)d_in[19];
    const float* qw2 = (const float*)d_in[20]; const float* qb2 = (const float*)d_in[21];

    float* p = (float*)d_ws;
    float* feat  = p; p += (size_t)B_ * CENC * H_ * W_;   // 0.5M
    float* col   = p; p += (size_t)NPIX * KCONV;          // 4.7M
    float* wtc   = p; p += (size_t)KCONV * HID;
    float* wtf   = p; p += (size_t)KCONV * HID;
    float* coefN = p; p += (size_t)NPIX * HID;
    float* freqN = p; p += (size_t)NPIX * HID;
    float* fmN   = p; p += (size_t)NPIX * HID;
    float* X     = p; p += (size_t)M4 * LDX;              // 34.6M
    float* Hbuf  = p; p += (size_t)M4 * HID;              // 33.5M
    float* Pk    = p; p += (size_t)M4 * HID;
    float* Pv    = p; p += (size_t)M4 * HID;
    float* Xattn = p; p += (size_t)BQ * HID;
    float* Hq    = p; p += (size_t)BQ * HID;

    k_conv_enc<<<2048, 256, 0, stream>>>(inp, enc_w, enc_b, feat);
    k_wt<<<576, 256, 0, stream>>>(coef_w, freq_w, wtc, wtf);
    k_im2col<<<18432, 256, 0, stream>>>(feat, col);

    // 64->256 convs as WMMA GEMMs (NHWC outputs)
    k_gemm<<<dim3(4, 128), 128, 0, stream>>>(col, KCONV, wtc, HID, coef_b,
                                             coefN, HID, nullptr, 0, KCONV, 0);
    k_gemm<<<dim3(4, 128), 128, 0, stream>>>(col, KCONV, wtf, HID, freq_b,
                                             freqN, HID, nullptr, 0, KCONV, 0);

    k_fmap<<<NPIX, 128, 0, stream>>>(coefN, freqN, fmN);
    k_prepare<<<M4, 128, 0, stream>>>(coord, cell, coefN, freqN, phase_w, X);

    // K path: Hbuf = relu(X@kw1+kb1); Pk = kv * (Hbuf@kw2+kb2)
    k_gemm<<<dim3(4, 2048), 128, 0, stream>>>(X, LDX, kw1, HID, kb1,
                                              Hbuf, HID, nullptr, 0, 260, 1);
    k_gemm<<<dim3(4, 2048), 128, 0, stream>>>(Hbuf, HID, kw2, HID, kb2,
                                              Pk, HID, X, LDX, HID, 0);
    // V path (reuse Hbuf)
    k_gemm<<<dim3(4, 2048), 128, 0, stream>>>(X, LDX, vw1, HID, vb1,
                                              Hbuf, HID, nullptr, 0, 260, 1);
    k_gemm<<<dim3(4, 2048), 128, 0, stream>>>(Hbuf, HID, vw2, HID, vb2,
                                              Pv, HID, X, LDX, HID, 0);

    k_attn<<<BQ / 8, 256, 0, stream>>>(coord, fmN, Pk, Pv, Xattn);

    // final MLP layer 1
    k_gemm<<<dim3(4, 512), 128, 0, stream>>>(Xattn, HID, qw1, HID, qb1,
                                             Hq, HID, nullptr, 0, HID, 1);
    k_final<<<BQ / 128, 128, 0, stream>>>(Hq, qw2, qb2, inp, coord, (float*)d_out);
}